// AdditiveScorer_6597069767500
// MI455X (gfx1250) — compile-verified
//
#include <hip/hip_runtime.h>
#include <stdint.h>

// Problem constants (match reference)
#define BB 2
#define NQ 1024
#define NK 1024
#define DD 512
#define HH 32

typedef __attribute__((ext_vector_type(2))) float v2f;
typedef __attribute__((ext_vector_type(8))) float v8f;

// ---------------------------------------------------------------------------
// gfx1250 async global->LDS DMA (ASYNCcnt-tracked), ISA ch.8 / 15.18.3.
// VDST = LDS byte address VGPR, VADDR = 32-bit global offset, SADDR = base.
// ---------------------------------------------------------------------------
__device__ __forceinline__ void async_ld_b128(uint32_t lds_addr, uint32_t voff,
                                              uint64_t sbase) {
    asm volatile("global_load_async_to_lds_b128 %0, %1, %2"
                 :: "v"(lds_addr), "v"(voff), "s"(sbase) : "memory");
}
__device__ __forceinline__ void wait_async0() {
    asm volatile("s_wait_asynccnt 0x0" ::: "memory");
}

// ---------------------------------------------------------------------------
// Phase 1: out[M x 32] = X[M x 512] * W[512 x 32]  (+ optional bias per column)
// One wave computes one 16-row M tile using V_WMMA_F32_16X16X4_F32.
// 8 waves / block, grid.x * 8 * 16 == M == 2048.
// ---------------------------------------------------------------------------
__global__ void proj_wmma_f32(const float* __restrict__ X,
                              const float* __restrict__ W,
                              const float* __restrict__ bias,
                              float* __restrict__ out,
                              int addBias) {
    const int lane = threadIdx.x & 31;
    const int wave = threadIdx.x >> 5;
    const int rowbase = (blockIdx.x * 8 + wave) * 16;

    const int r16  = lane & 15;        // M row (A) / N col (B,C,D) within tile
    const int hi   = lane >> 4;        // half-wave select
    const int koff = hi * 2;           // lanes 0-15 hold K=0,1; 16-31 hold K=2,3

    v8f c0 = {};   // columns 0..15
    v8f c1 = {};   // columns 16..31

    const float* Arow = X + (size_t)(rowbase + r16) * DD;

    for (int kk = 0; kk < DD; kk += 4) {
        // A fragment: 16x4 f32 (ISA 7.12.2 layout)
        const float2 av = *(const float2*)(Arow + kk + koff);
        v2f a; a.x = av.x; a.y = av.y;

        // B fragment: 4x16 f32, lane holds column r16(+16), K pair per half-wave
        const float* Wr0 = W + (size_t)(kk + koff) * HH;   // row K = kk+koff
        const float* Wr1 = Wr0 + HH;                       // row K = kk+koff+1
        v2f b0, b1v;
        b0.x  = Wr0[r16];       b0.y  = Wr1[r16];
        b1v.x = Wr0[r16 + 16];  b1v.y = Wr1[r16 + 16];

        c0 = __builtin_amdgcn_wmma_f32_16x16x4_f32(false, a, false, b0,
                                                   (short)0, c0, false, false);
        c1 = __builtin_amdgcn_wmma_f32_16x16x4_f32(false, a, false, b1v,
                                                   (short)0, c1, false, false);
    }

    if (addBias) {  // fold b1 into the query projection once
        const float bA = bias[r16];
        const float bB = bias[r16 + 16];
        #pragma unroll
        for (int r = 0; r < 8; ++r) { c0[r] += bA; c1[r] += bB; }
    }

    // C/D layout: VGPR r -> row r + 8*hi, lane%16 -> column
    #pragma unroll
    for (int r = 0; r < 8; ++r) {
        const int row = rowbase + r + 8 * hi;
        out[(size_t)row * HH + r16]      = c0[r];
        out[(size_t)row * HH + 16 + r16] = c1[r];
    }
}

// ---------------------------------------------------------------------------
// Phase 2: logits[b,q,k] = W2 . tanh(hk[b,k,:] + hqb[b,q,:]) + b2
// (b1 already folded into hqb). Block tile: 8 q (one per wave) x 64 k
// (each lane: k=lane and k=lane+32 -> 8 independent tanhs per TRANS group).
// Tiles staged via gfx1250 global_load_async_to_lds_b128 (ASYNCcnt).
// ---------------------------------------------------------------------------
#define TQ 8
#define TK 64
#define HKPAD 36   // 144B row pitch: 16B aligned, spreads LDS banks

__global__ void additive_score_kernel(const float* __restrict__ hk,
                                      const float* __restrict__ hqb,
                                      const float* __restrict__ W2,
                                      const float* __restrict__ b2,
                                      float* __restrict__ out) {
    __shared__ float hkS[TK * HKPAD];
    __shared__ float hqS[TQ * HH];
    __shared__ float w2S[HH];

    const int tid = threadIdx.x;           // 256 threads = 8 waves
    const int b   = blockIdx.z;
    const int qb  = blockIdx.y * TQ;
    const int kb  = blockIdx.x * TK;

    const uint32_t hkS_base = (uint32_t)(uintptr_t)&hkS[0];
    const uint32_t hqS_base = (uint32_t)(uintptr_t)&hqS[0];
    const uint32_t w2S_base = (uint32_t)(uintptr_t)&w2S[0];

    // hk tile: 64 rows x 32 f32 = 512 float4 chunks; 2 per thread (DMA to LDS)
    {
        const uint64_t sbase = (uint64_t)hk;
        #pragma unroll
        for (int i = 0; i < 2; ++i) {
            const int idx128 = tid + i * 256;           // 0..511
            const int r  = idx128 >> 3;                 // k row in tile
            const int h4 = (idx128 & 7) * 4;            // h start
            const uint32_t voff =
                (uint32_t)((((size_t)b * NK + kb + r) * HH + h4) * sizeof(float));
            const uint32_t ldst =
                hkS_base + (uint32_t)((r * HKPAD + h4) * sizeof(float));
            async_ld_b128(ldst, voff, sbase);
        }
    }
    // hq tile: 8 rows x 32 f32 = 64 float4 chunks (rows contiguous in LDS)
    if (tid < 64) {
        const int r  = tid >> 3;
        const int h4 = (tid & 7) * 4;
        const uint32_t voff =
            (uint32_t)((((size_t)b * NQ + qb + r) * HH + h4) * sizeof(float));
        async_ld_b128(hqS_base + (uint32_t)((r * HH + h4) * sizeof(float)),
                      voff, (uint64_t)hqb);
    }
    // W2: 8 float4 chunks
    if (tid < 8) {
        async_ld_b128(w2S_base + (uint32_t)(tid * 16),
                      (uint32_t)(tid * 16), (uint64_t)W2);
    }
    wait_async0();        // own ASYNCcnt drained before signaling barrier
    __syncthreads();      // all waves' DMAs visible in LDS

    const int lane = tid & 31;             // k within tile (and k+32)
    const int wave = tid >> 5;             // q within tile
    const float* hkr0 = &hkS[lane * HKPAD];
    const float* hkr1 = &hkS[(lane + 32) * HKPAD];
    const float* hqr  = &hqS[wave * HH];

    float acc0 = 0.f, acc1 = 0.f;
    #pragma unroll
    for (int h = 0; h < HH; h += 4) {
        const float4 a0 = *(const float4*)(hkr0 + h);  // ds_load_b128
        const float4 a1 = *(const float4*)(hkr1 + h);
        const float4 aq = *(const float4*)(hqr + h);   // broadcast row
        const float4 w  = *(const float4*)(w2S + h);

        const float x0 = a0.x + aq.x, x1 = a0.y + aq.y;
        const float x2 = a0.z + aq.z, x3 = a0.w + aq.w;
        const float y0 = a1.x + aq.x, y1 = a1.y + aq.y;
        const float y2 = a1.z + aq.z, y3 = a1.w + aq.w;

        float t0, t1, t2, t3, u0, u1, u2, u3;
        // 8 pipelined TRANS ops; trailing s_delay_alu waits on the newest
        // TRANS -> all earlier ones complete (in-order TRANS completion),
        // covering the CDNA5 multicycle co-execution hazard in one stall.
        asm("v_tanh_f32 %0, %8\n\t"
            "v_tanh_f32 %1, %9\n\t"
            "v_tanh_f32 %2, %10\n\t"
            "v_tanh_f32 %3, %11\n\t"
            "v_tanh_f32 %4, %12\n\t"
            "v_tanh_f32 %5, %13\n\t"
            "v_tanh_f32 %6, %14\n\t"
            "v_tanh_f32 %7, %15\n\t"
            "s_delay_alu instid0(TRANS32_DEP_1)"
            : "=v"(t0), "=v"(t1), "=v"(t2), "=v"(t3),
              "=v"(u0), "=v"(u1), "=v"(u2), "=v"(u3)
            : "v"(x0), "v"(x1), "v"(x2), "v"(x3),
              "v"(y0), "v"(y1), "v"(y2), "v"(y3));

        acc0 += w.x * t0 + w.y * t1 + w.z * t2 + w.w * t3;
        acc1 += w.x * u0 + w.y * u1 + w.z * u2 + w.w * u3;
    }

    // Coalesced: lane-contiguous along k
    const float bias2 = b2[0];
    float* orow = out + ((size_t)b * NQ + qb + wave) * NK + kb;
    orow[lane]      = acc0 + bias2;
    orow[lane + 32] = acc1 + bias2;
}

// ---------------------------------------------------------------------------
// Launch
// ---------------------------------------------------------------------------
extern "C" void kernel_launch(void* const* d_in, const int* in_sizes, int n_in,
                              void* d_out, int out_size, void* d_ws, size_t ws_size,
                              hipStream_t stream) {
    (void)in_sizes; (void)n_in; (void)out_size; (void)ws_size;

    const float* keys    = (const float*)d_in[0];  // (B, NK, D)
    const float* queries = (const float*)d_in[1];  // (B, NQ, D)
    const float* W1      = (const float*)d_in[2];  // (2D, H) row-major
    const float* b1      = (const float*)d_in[3];  // (H)
    const float* W2      = (const float*)d_in[4];  // (H, 1)
    const float* b2      = (const float*)d_in[5];  // (1)

    float* hk  = (float*)d_ws;                       // 2048*32 f32 = 256 KB
    float* hqb = hk + (size_t)BB * NK * HH;          // 2048*32 f32 = 256 KB

    // M = B*NK = B*NQ = 2048 rows -> 16 blocks * 8 waves * 16 rows
    proj_wmma_f32<<<dim3(16), 256, 0, stream>>>(keys, W1, b1, hk, 0);
    proj_wmma_f32<<<dim3(16), 256, 0, stream>>>(queries, W1 + (size_t)DD * HH,
                                                b1, hqb, 1);

    dim3 grid(NK / TK, NQ / TQ, BB);                 // (16, 128, 2)
    additive_score_kernel<<<grid, 256, 0, stream>>>(hk, hqb, W2, b2,
                                                    (float*)d_out);
}